// ProtoICLHead_16441134809588
// MI455X (gfx1250) — compile-verified
//
#include <hip/hip_runtime.h>
#include <hip/hip_bf16.h>

typedef __attribute__((ext_vector_type(2))) float v2f;
typedef __attribute__((ext_vector_type(8))) float v8f;
typedef __attribute__((ext_vector_type(4))) unsigned int v4u;
typedef __attribute__((ext_vector_type(4))) int v4i;
typedef __attribute__((ext_vector_type(8))) int v8i;

#define D_DIM 128
#define C_DIM 256
#define EPSF 1e-8f
#define NEG_FILLV -1.0e6f
#define CHUNK 64   // prototype columns staged in LDS at a time
#define LROW 132   // dwords per proto row in LDS: 128 + 4 pad (bank spread)

__device__ __forceinline__ float wave_reduce_sum(float x) {
#pragma unroll
  for (int off = 16; off >= 1; off >>= 1) x += __shfl_xor(x, off, 32);
  return x;
}

// ---------------------------------------------------------------------------
// Kernel 1: one wave per class. Deterministic prototype means (no atomics):
// ballot-scan of labels in ascending row order; each lane owns 4 dims.
// ---------------------------------------------------------------------------
__global__ __launch_bounds__(32) void proto_build(
    const float* __restrict__ sfeat, const int* __restrict__ slab,
    float* __restrict__ protos, float* __restrict__ psq,
    float* __restrict__ cnts, int K) {
  const int c = blockIdx.x;
  const int lane = threadIdx.x;
  float a0 = 0.f, a1 = 0.f, a2 = 0.f, a3 = 0.f;
  int cnt = 0;
  for (int base = 0; base < K; base += 32) {
    const int lab = slab[base + lane];
    unsigned long long m = __ballot(lab == c);
    while (m) {
      const int j = __ffsll(m) - 1;
      m &= (m - 1);
      const float4 v =
          *(const float4*)(sfeat + (size_t)(base + j) * D_DIM + 4 * lane);
      float ss = v.x * v.x + v.y * v.y + v.z * v.z + v.w * v.w;
      ss = wave_reduce_sum(ss);  // full ||row||^2 (fixed tree: deterministic)
      const float rn = 1.0f / fmaxf(sqrtf(ss), EPSF);
      a0 += v.x * rn; a1 += v.y * rn; a2 += v.z * rn; a3 += v.w * rn;
      ++cnt;
    }
  }
  const float fc = (float)cnt;
  const float inv = 1.0f / fmaxf(fc, 1.0f);
  float4 p;
  p.x = a0 * inv; p.y = a1 * inv; p.z = a2 * inv; p.w = a3 * inv;
  *(float4*)(protos + (size_t)c * D_DIM + 4 * lane) = p;
  float pp = p.x * p.x + p.y * p.y + p.z * p.z + p.w * p.w;
  pp = wave_reduce_sum(pp);
  if (lane == 0) { psq[c] = pp; cnts[c] = fc; }
}

// ---------------------------------------------------------------------------
// Kernel 2: logits via V_WMMA_F32_16X16X4_F32, protos staged by the Tensor
// Data Mover with hardware pad insertion (pad_interval=128 DW, pad_amount=4 DW
// -> LDS rows of 132 dwords; lane stride 132 = 4 mod 64 banks: conflict-free).
// Block = 128 threads (4 waves). Wave w owns rows m0..m0+15 and all 256 cols.
// A operand (16x4 f32): lane l = row (l&15); VGPR0/1 hold K = 4s+2h+{0,1},
// h = l>>4 — q strip preloaded & normalized into registers in this layout.
// B operand mirrors it: lane l = col (l&15), same K pairs, read as ds_load_b64.
// ---------------------------------------------------------------------------
__global__ __launch_bounds__(128) void proto_logits(
    const float* __restrict__ qfeat, const float* __restrict__ protos,
    const float* __restrict__ psq, const float* __restrict__ cnts,
    float* __restrict__ out) {
  __shared__ __align__(16) float bsh[CHUNK * LROW];  // 33792 B

  const int tid = threadIdx.x;
  const int wave = tid >> 5;
  const int lane = tid & 31;
  const int lrow = lane & 15;
  const int h = lane >> 4;

  const int m0 = blockIdx.x * 64 + wave * 16;
  const float* qrow = qfeat + (size_t)(m0 + lrow) * D_DIM;

  // --- preload + L2-normalize this wave's 16 q rows into A-operand regs ---
  float2 areg[32];
  float ss = 0.f;
#pragma unroll
  for (int s = 0; s < 32; ++s) {
    const float2 v = *(const float2*)(qrow + 4 * s + 2 * h);
    areg[s] = v;
    ss += v.x * v.x + v.y * v.y;
  }
  ss += __shfl_xor(ss, 16, 32);  // combine the two half-rows
  const float rn = 1.0f / fmaxf(sqrtf(ss), EPSF);
  const float qsq = ss * rn * rn;  // ||q_norm||^2, matches reference q_sq
#pragma unroll
  for (int s = 0; s < 32; ++s) { areg[s].x *= rn; areg[s].y *= rn; }

  for (int ch = 0; ch < C_DIM / CHUNK; ++ch) {
    const int c0 = ch * CHUNK;
    __syncthreads();  // previous chunk's LDS reads done

#if __has_builtin(__builtin_amdgcn_tensor_load_to_lds)
    // --- TDM: each wave DMAs 16 proto rows (128 f32) into padded LDS ---
    {
      const unsigned int lds_base = (unsigned int)__builtin_amdgcn_readfirstlane(
          (int)((unsigned int)(uintptr_t)bsh + (unsigned)wave * 16u * LROW * 4u));
      const unsigned long long ga =
          (unsigned long long)(uintptr_t)(protos +
                                          (size_t)(c0 + wave * 16) * D_DIM);
      const unsigned int ga_lo =
          (unsigned int)__builtin_amdgcn_readfirstlane((int)(unsigned int)ga);
      const unsigned int ga_hi = (unsigned int)__builtin_amdgcn_readfirstlane(
          (int)((unsigned int)(ga >> 32) & 0x01FFFFFFu));
      // D# group0: count=1 | lds_addr | global_addr[56:0] | type=2
      v4u g0;
      g0.x = 1u;                         // count=1 valid descriptor
      g0.y = lds_base;                   // bits [63:32] lds_addr
      g0.z = ga_lo;                      // bits [95:64] global_addr lo
      g0.w = ga_hi | (2u << 30);         // global_addr hi | type=2 (image)
      // D# group1: data_size=4B(2), pad_enable, pad_interval=6 (128 DW),
      // pad_amount=3 (4 DW); tensor_dim0=128, tensor_dim1=16,
      // tile_dim0=128, tile_dim1=16, tile_dim2=0, dim0_stride=128.
      v8i g1;
      g1.s0 = (int)((2u << 16) | (1u << 20) | (6u << 22) | (3u << 25));
      g1.s1 = (int)((128u & 0xFFFFu) << 16);              // tensor_dim0 lo16
      g1.s2 = (int)((128u >> 16) | ((16u & 0xFFFFu) << 16)); // dim0 hi|dim1 lo
      g1.s3 = (int)((16u >> 16) | (128u << 16));          // dim1 hi | tile_dim0
      g1.s4 = (int)(16u);                                 // tile_dim1 | tile_dim2=0
      g1.s5 = (int)(128u);                                // tensor_dim0_stride lo
      g1.s6 = 0;                                          // stride hi | dim1_stride lo
      g1.s7 = 0;
      const v4i z4 = {0, 0, 0, 0};
#if defined(__clang_major__) && (__clang_major__ >= 23)
      const v8i z8 = {0, 0, 0, 0, 0, 0, 0, 0};
      __builtin_amdgcn_tensor_load_to_lds(g0, g1, z4, z4, z8, 0);
#else
      __builtin_amdgcn_tensor_load_to_lds(g0, g1, z4, z4, 0);
#endif
      __builtin_amdgcn_s_wait_tensorcnt(0);
    }
#else
    // --- fallback: manual coalesced staging into the same padded layout ---
    for (int i = tid; i < CHUNK * 32; i += 128) {
      const int col = i >> 5;
      const int j = i & 31;
      const float4 v =
          *(const float4*)(protos + (size_t)(c0 + col) * D_DIM + 4 * j);
      *(float4*)&bsh[col * LROW + 4 * j] = v;  // 16B aligned: 528*col + 16*j
    }
#endif
    __syncthreads();

    v8f acc[4];
#pragma unroll
    for (int t = 0; t < 4; ++t)
      acc[t] = (v8f){0.f, 0.f, 0.f, 0.f, 0.f, 0.f, 0.f, 0.f};

#pragma unroll
    for (int s = 0; s < 32; ++s) {  // K advances by 4 per step
      const v2f A0 = {areg[s].x, areg[s].y};
#pragma unroll
      for (int t = 0; t < 4; ++t) {
        const float2 b =
            *(const float2*)&bsh[(t * 16 + lrow) * LROW + 4 * s + 2 * h];
        const v2f B0 = {b.x, b.y};
        acc[t] = __builtin_amdgcn_wmma_f32_16x16x4_f32(
            false, A0, false, B0, (short)0, acc[t], false, false);
      }
    }

    // --- epilogue: logits = 4*dot - 2*(q_sq + p_sq); absent -> -1e6 ---
#pragma unroll
    for (int t = 0; t < 4; ++t) {
      const int colg = c0 + t * 16 + lrow;
      const float ps = psq[colg];
      const bool present = cnts[colg] > 0.0f;
#pragma unroll
      for (int r = 0; r < 8; ++r) {
        // C/D layout: VGPR r, lane l -> M = r + 8*(l>>4), N = l&15
        const float qs = __shfl(qsq, r + 8 * h, 32);
        float val = 4.0f * acc[t][r] - 2.0f * (qs + ps);
        if (!present) val = NEG_FILLV;
        out[(size_t)(m0 + r + 8 * h) * C_DIM + colg] = val;
      }
    }
  }
}

extern "C" void kernel_launch(void* const* d_in, const int* in_sizes, int n_in,
                              void* d_out, int out_size, void* d_ws,
                              size_t ws_size, hipStream_t stream) {
  const float* sfeat = (const float*)d_in[0];
  const int* slab = (const int*)d_in[1];
  const float* qfeat = (const float*)d_in[2];
  // d_in[3] = num_classes (device scalar); C=256 fixed by the problem.
  const int K = in_sizes[1];
  const int M = in_sizes[2] / D_DIM;

  float* protos = (float*)d_ws;                 // [C, D]
  float* psq = protos + (size_t)C_DIM * D_DIM;  // [C]
  float* cnts = psq + C_DIM;                    // [C]

  proto_build<<<dim3(C_DIM), dim3(32), 0, stream>>>(sfeat, slab, protos, psq,
                                                    cnts, K);
  proto_logits<<<dim3(M / 64), dim3(128), 0, stream>>>(qfeat, protos, psq,
                                                       cnts, (float*)d_out);
}